// StateExpDecay_85461259256063
// MI455X (gfx1250) — compile-verified
//
#include <hip/hip_runtime.h>

// CDNA5 / gfx1250, wave32. Fused StateExpDecay:
//   out = (Qinv * diag(exp(-(T/tau)*|S|)) * Q) @ x    per pixel, C=64.
// Kernel 1 builds the fused 64x64 matrix M, kernel 2 is an M(64x64) x X(64xN)
// GEMM on the f32 WMMA pipe, N = B*H*W = 1,048,576.

typedef __attribute__((ext_vector_type(2))) float v2f;
typedef __attribute__((ext_vector_type(8))) float v8f;

#define C_DIM 64
#define HW_DIM (512 * 512)
#define B_DIM 4
#define TAU 0.5f
#define LDS_STRIDE 66   // 64 + 2 pad dwords: de-conflicts strided ds_load_b64

// ---------------------------------------------------------------------------
// Kernel 1: M[f][c] = sum_k Qinv[f][k] * exp(-(T/tau)*|S[k]|) * Q[k][c]
// 64 blocks x 64 threads; ~0.5 MFLOP, negligible.
// ---------------------------------------------------------------------------
__global__ void build_fused_matrix(const float* __restrict__ S,
                                   const float* __restrict__ Q,
                                   const float* __restrict__ Qinv,
                                   const int* __restrict__ T,
                                   float* __restrict__ M) {
  const int f = blockIdx.x;   // output row 0..63
  const int c = threadIdx.x;  // output col 0..63
  const float tscale = -(1.0f / TAU) * (float)T[0];  // = -2*T
  float acc = 0.0f;
  for (int k = 0; k < C_DIM; ++k) {
    const float d = __expf(tscale * fabsf(S[k]));
    acc += Qinv[f * C_DIM + k] * d * Q[k * C_DIM + c];
  }
  M[f * C_DIM + c] = acc;
}

// ---------------------------------------------------------------------------
// Kernel 2: out[b][f][p] = sum_c M[f][c] * x[b][c][p]
// Block = 256 threads = 8 waves. Each wave owns 16 contiguous pixels and
// produces all 64 output channels for them via v_wmma_f32_16x16x4_f32.
//
// WMMA f32 16x16x4 VGPR layouts (ISA 7.12.2):
//   A (16x4, M in lanes): lane&15 = row M; VGPR j, half h -> K = 2h + j
//   B (4x16, N in lanes): lane&15 = col N; VGPR j, half h -> K = 2h + j
//   C/D (16x16):          lane&15 = col N; VGPR r, half h -> M = r + 8h
// ---------------------------------------------------------------------------
__global__ void state_expdecay_gemm(const float* __restrict__ x,
                                    const float* __restrict__ M,
                                    float* __restrict__ out) {
  __shared__ __attribute__((aligned(16))) float ldsM[C_DIM * LDS_STRIDE];

  // Stage M (16 KB) into LDS once per block, padded rows.
  for (int i = threadIdx.x; i < C_DIM * C_DIM; i += blockDim.x) {
    const int f = i >> 6;
    const int k = i & 63;
    ldsM[f * LDS_STRIDE + k] = M[i];
  }
  __syncthreads();

  const int wave = threadIdx.x >> 5;   // 0..7
  const int lane = threadIdx.x & 31;
  const int hi   = lane >> 4;          // half-wave: 0 or 1
  const int col  = lane & 15;          // pixel within the 16-wide column / row M for A

  const long long b  = blockIdx.y;
  const long long n0 = ((long long)blockIdx.x * 8 + wave) * 16;  // first pixel

  const float* __restrict__ xcol = x + b * C_DIM * HW_DIM + n0 + col;
  float* __restrict__ ocol       = out + b * C_DIM * HW_DIM + n0 + col;

  v8f acc0 = {}, acc1 = {}, acc2 = {}, acc3 = {};

#pragma unroll
  for (int k0 = 0; k0 < C_DIM; k0 += 4) {
    const int kb = k0 + 2 * hi;  // K pair handled by this half-wave

    // B tile: x[kb..kb+1][n0+col] — two coalesced 64B segments per load.
    v2f bvec;
    bvec.x = xcol[(long long)(kb + 0) * HW_DIM];
    bvec.y = xcol[(long long)(kb + 1) * HW_DIM];

    // A tiles for the 4 output-channel row-tiles: ds_load_b64 each
    // (kb is even and rows are 8B-aligned: 66*4 = 264 bytes).
    const v2f a0 = *(const v2f*)&ldsM[(col +  0) * LDS_STRIDE + kb];
    const v2f a1 = *(const v2f*)&ldsM[(col + 16) * LDS_STRIDE + kb];
    const v2f a2 = *(const v2f*)&ldsM[(col + 32) * LDS_STRIDE + kb];
    const v2f a3 = *(const v2f*)&ldsM[(col + 48) * LDS_STRIDE + kb];

    acc0 = __builtin_amdgcn_wmma_f32_16x16x4_f32(false, a0, false, bvec,
                                                 (short)0, acc0, false, false);
    acc1 = __builtin_amdgcn_wmma_f32_16x16x4_f32(false, a1, false, bvec,
                                                 (short)0, acc1, false, false);
    acc2 = __builtin_amdgcn_wmma_f32_16x16x4_f32(false, a2, false, bvec,
                                                 (short)0, acc2, false, false);
    acc3 = __builtin_amdgcn_wmma_f32_16x16x4_f32(false, a3, false, bvec,
                                                 (short)0, acc3, false, false);
  }

  // Store D tiles: element r of acc_t -> channel f = t*16 + r + 8*hi, pixel n0+col.
#pragma unroll
  for (int r = 0; r < 8; ++r) {
    const long long fr = r + 8 * hi;
    ocol[(fr +  0) * HW_DIM] = acc0[r];
    ocol[(fr + 16) * HW_DIM] = acc1[r];
    ocol[(fr + 32) * HW_DIM] = acc2[r];
    ocol[(fr + 48) * HW_DIM] = acc3[r];
  }
}

// ---------------------------------------------------------------------------
extern "C" void kernel_launch(void* const* d_in, const int* in_sizes, int n_in,
                              void* d_out, int out_size, void* d_ws, size_t ws_size,
                              hipStream_t stream) {
  const float* x    = (const float*)d_in[0];
  const float* S    = (const float*)d_in[1];
  const float* Q    = (const float*)d_in[2];
  const float* Qinv = (const float*)d_in[3];
  const int*   T    = (const int*)d_in[4];
  float* outp = (float*)d_out;
  float* Mbuf = (float*)d_ws;  // 64*64 f32 = 16 KB scratch

  build_fused_matrix<<<dim3(C_DIM), dim3(C_DIM), 0, stream>>>(S, Q, Qinv, T, Mbuf);

  // 8 waves/block * 16 pixels/wave = 128 pixels per block.
  dim3 grid(HW_DIM / 128, B_DIM);
  state_expdecay_gemm<<<grid, dim3(256), 0, stream>>>(x, Mbuf, outp);
}